// SANWrapper_49881750176303
// MI455X (gfx1250) — compile-verified
//
#include <hip/hip_runtime.h>

#define N_NODES 100000
#define N_EDGES 1600000
#define D 48
#define EDGES_PER_BLOCK 128   // 8 waves x 16 edges
#define THREADS 256
#define NPAIRS (D / 2)        // 24 K-pairs

typedef __attribute__((ext_vector_type(2))) float v2f;
typedef __attribute__((ext_vector_type(8))) float v8f;

__global__ void san_zero_kernel(float* __restrict__ out, int n4) {
    int i = blockIdx.x * blockDim.x + threadIdx.x;
    int stride = gridDim.x * blockDim.x;
    float4* o4 = reinterpret_cast<float4*>(out);
    float4 z; z.x = 0.f; z.y = 0.f; z.z = 0.f; z.w = 0.f;
    for (; i < n4; i += stride) o4[i] = z;
}

__global__ __launch_bounds__(THREADS)
void san_fused_wmma_kernel(const float* __restrict__ x1,
                           const float* __restrict__ Wm,
                           const int* __restrict__ esrc,
                           const int* __restrict__ etgt,
                           float* __restrict__ out) {
    // W staged K-pair-interleaved: sWp[p*2*D + n*2 + j] = W[(2p+j)*D + n]
    // so a 4x16 B fragment's two K values per lane are one contiguous v2f.
    __shared__ float sWp[NPAIRS * 2 * D];   // 9216 B
    __shared__ int sSrc[EDGES_PER_BLOCK];
    __shared__ int sTgt[EDGES_PER_BLOCK];

    const int tid = threadIdx.x;
    const int blockEdge0 = blockIdx.x * EDGES_PER_BLOCK;

    #pragma unroll
    for (int i = tid; i < NPAIRS * 2 * D; i += THREADS) {
        const int p   = i / (2 * D);
        const int rem = i - p * (2 * D);
        const int n   = rem >> 1;
        const int j   = rem & 1;
        sWp[i] = Wm[(2 * p + j) * D + n];
    }
    if (tid < EDGES_PER_BLOCK) {
        sSrc[tid] = esrc[blockEdge0 + tid];
        sTgt[tid] = etgt[blockEdge0 + tid];
    }
    __syncthreads();

    const int wave = tid >> 5;       // wave32
    const int lane = tid & 31;
    const int g    = lane >> 4;      // lane half: 0 or 1
    const int lm   = lane & 15;      // M (for A) / N (for B,C) within half
    const int waveEdge0 = wave * 16;

    // A-matrix row this lane feeds (M = lm in both halves).
    const float* __restrict__ arow =
        x1 + (size_t)(blockEdge0 + waveEdge0 + lm) * D;

    v8f acc[3];
    acc[0] = v8f{}; acc[1] = v8f{}; acc[2] = v8f{};

    #pragma unroll
    for (int k0 = 0; k0 < D; k0 += 4) {
        // A frag (16x4 f32): lane half g covers K = k0+2g, k0+2g+1 -> one b64
        const v2f a = *reinterpret_cast<const v2f*>(arow + k0 + 2 * g);

        const int p = (k0 >> 1) + g;          // K-pair index for this lane half
        const float* __restrict__ brow = &sWp[p * 2 * D];

        #pragma unroll
        for (int nt = 0; nt < 3; ++nt) {
            // B frag (4x16 f32): contiguous pair -> one ds_load_b64
            const v2f b = *reinterpret_cast<const v2f*>(brow + 2 * (nt * 16 + lm));
            acc[nt] = __builtin_amdgcn_wmma_f32_16x16x4_f32(
                /*neg_a=*/false, a, /*neg_b=*/false, b,
                /*c_mod=*/(short)0, acc[nt],
                /*reuse_a=*/false, /*reuse_b=*/false);
        }
    }

    // Scatter: C/D layout -> vgpr r holds row M = r + 8*g, col N = lm.
    #pragma unroll
    for (int nt = 0; nt < 3; ++nt) {
        const int col = nt * 16 + lm;
        #pragma unroll
        for (int r = 0; r < 8; ++r) {
            const int eLocal = waveEdge0 + r + 8 * g;
            const float v = acc[nt][r];
            const int t = sTgt[eLocal];
            const int s = sSrc[eLocal];
            atomicAdd(&out[(size_t)t * D + col], v);
            atomicAdd(&out[(size_t)s * D + col], -v);
        }
    }
}

extern "C" void kernel_launch(void* const* d_in, const int* in_sizes, int n_in,
                              void* d_out, int out_size, void* d_ws, size_t ws_size,
                              hipStream_t stream) {
    const float* x1   = (const float*)d_in[0];   // [N_EDGES, D] f32
    const float* Wm   = (const float*)d_in[1];   // [D, D] f32
    const int*   esrc = (const int*)d_in[2];     // [N_EDGES] i32
    const int*   etgt = (const int*)d_in[3];     // [N_EDGES] i32
    float* out = (float*)d_out;                  // [N_NODES, D] f32

    const int n4 = (N_NODES * D) / 4;
    san_zero_kernel<<<2048, THREADS, 0, stream>>>(out, n4);

    const int blocks = N_EDGES / EDGES_PER_BLOCK;  // 12500
    san_fused_wmma_kernel<<<blocks, THREADS, 0, stream>>>(x1, Wm, esrc, etgt, out);
}